// MSGDA_70858370450103
// MI455X (gfx1250) — compile-verified
//
#include <hip/hip_runtime.h>
#include <hip/hip_bf16.h>
#include <math.h>
#include <stdint.h>

#define S_DOM 3
#define NNODE 4096
#define NEDGE 65536
#define HDIM 128
#define SN (S_DOM * NNODE)      // 12288
#define KNN 5
#define BN_EPS 1e-5f

typedef __attribute__((ext_vector_type(2))) float v2f;
typedef __attribute__((ext_vector_type(4))) float v4f;
typedef __attribute__((ext_vector_type(8))) float v8f;
typedef __attribute__((ext_vector_type(4))) int   v4i;

// ---- CDNA5 async global->LDS copy (ASYNCcnt-tracked), with sync fallback ----
#if defined(__has_builtin)
#  if __has_builtin(__builtin_amdgcn_global_load_async_to_lds_b128)
#    define HAVE_ASYNC_LDS 1
#  endif
#endif
#ifndef HAVE_ASYNC_LDS
#  define HAVE_ASYNC_LDS 0
#endif

__device__ __forceinline__ void async_copy16(const float* g, float* l) {
#if HAVE_ASYNC_LDS
    __builtin_amdgcn_global_load_async_to_lds_b128(
        (__attribute__((address_space(1))) v4i*)(uintptr_t)g,
        (__attribute__((address_space(3))) v4i*)(unsigned)(uintptr_t)l,
        0, 0);
#else
    *(v4f*)l = *(const v4f*)g;
#endif
}

__device__ __forceinline__ void wait_async_all() {
#if HAVE_ASYNC_LDS
#  if defined(__has_builtin) && __has_builtin(__builtin_amdgcn_s_wait_asynccnt)
    __builtin_amdgcn_s_wait_asynccnt(0);
#  else
    asm volatile("s_wait_asynccnt 0x0" ::: "memory");
#  endif
#endif
}

// ---------------------------------------------------------------------------
// WMMA fp32 GEMM: C[M,128] = A[M,128] @ W[128,128] (+bias) (+relu)
// one wave (32 lanes) per 16x16 output tile; 32x chained V_WMMA_F32_16X16X4_F32
// ---------------------------------------------------------------------------
__global__ __launch_bounds__(32) void k_gemm128(const float* __restrict__ A,
                                                const float* __restrict__ W,
                                                const float* __restrict__ bias,
                                                int relu,
                                                float* __restrict__ C) {
    const int lane = threadIdx.x;
    const int half = lane >> 4;
    const int l16  = lane & 15;
    const int mt = blockIdx.x;
    const int nt = blockIdx.y;
    v8f acc = {};
#pragma unroll
    for (int ks = 0; ks < 32; ++ks) {
        const int k0 = ks * 4 + half * 2;
        v2f a, b;
        const float* ap = A + (mt * 16 + l16) * HDIM + k0;
        a.x = ap[0];
        a.y = ap[1];
        const float* bp = W + k0 * HDIM + nt * 16 + l16;
        b.x = bp[0];
        b.y = bp[HDIM];
        acc = __builtin_amdgcn_wmma_f32_16x16x4_f32(false, a, false, b,
                                                    (short)0, acc, false, false);
    }
#pragma unroll
    for (int r = 0; r < 8; ++r) {
        const int m = mt * 16 + r + 8 * half;
        const int n = nt * 16 + l16;
        float val = acc[r];
        if (bias) val += bias[n];
        if (relu) val = fmaxf(val, 0.0f);
        C[m * HDIM + n] = val;
    }
}

// ---------------------------------------------------------------------------
// Cosine-similarity (v @ v^T) with intra-domain masking and streaming top-5.
// 4 waves per workgroup (64 rows); the 16x128 B column tile is staged into a
// double-buffered, bank-padded LDS buffer via async global->LDS copies and
// shared by all 4 waves (4x less L2 traffic, coalesced b128 staging).
// Per tile: 32 chained fp32 WMMAs; 16x16 result bounced via LDS so lanes 0..15
// do an index-ascending strict-'>' top-5 insert (matches lax.top_k ties).
// ---------------------------------------------------------------------------
#define TWAVES 4
#define PADK 132   // LDS row stride (floats): 16 lanes -> 16 distinct banks

__device__ __forceinline__ void stage_tile(const float* __restrict__ v, int colbase,
                                           float* __restrict__ dst, int tid) {
#pragma unroll
    for (int c = tid; c < 512; c += 32 * TWAVES) {   // 512 chunks of 4 floats
        const int row = c >> 5;
        const int kk  = (c & 31) << 2;
        const float* g = v + (size_t)(colbase + row) * HDIM + kk;
        float* l = dst + row * PADK + kk;
        async_copy16(g, l);
    }
}

__global__ __launch_bounds__(32 * TWAVES) void k_simtopk(const float* __restrict__ v,
                                                         int* __restrict__ topk) {
    __shared__ float bbuf[2][16 * PADK];
    __shared__ float ctile[TWAVES][16 * 16];
    const int tid  = threadIdx.x;
    const int wave = tid >> 5;
    const int lane = tid & 31;
    const int half = lane >> 4;
    const int l16  = lane & 15;
    const int rowbase = blockIdx.x * (16 * TWAVES) + wave * 16;

    v2f a[32];
#pragma unroll
    for (int ks = 0; ks < 32; ++ks) {
        const int k0 = ks * 4 + half * 2;
        const float* p = v + (size_t)(rowbase + l16) * HDIM + k0;
        a[ks].x = p[0];
        a[ks].y = p[1];
    }

    float bv[KNN];
    int   bi[KNN];
#pragma unroll
    for (int p = 0; p < KNN; ++p) { bv[p] = -3.0e38f; bi[p] = 0; }

    const int rowdom = rowbase >> 12;   // 4096 nodes per domain; tiles aligned

    stage_tile(v, 0, bbuf[0], tid);
    wait_async_all();
    __syncthreads();

    for (int ct = 0; ct < SN / 16; ++ct) {
        const int cur = ct & 1;
        const int colbase = ct * 16;
        if (ct + 1 < SN / 16)
            stage_tile(v, colbase + 16, bbuf[1 - cur], tid);

        v8f acc = {};
#pragma unroll
        for (int ks = 0; ks < 32; ++ks) {
            const int k0 = ks * 4 + half * 2;
            const float* bp = &bbuf[cur][l16 * PADK + k0];
            v2f b;
            b.x = bp[0];
            b.y = bp[1];
            acc = __builtin_amdgcn_wmma_f32_16x16x4_f32(false, a[ks], false, b,
                                                        (short)0, acc, false, false);
        }
#pragma unroll
        for (int r = 0; r < 8; ++r)
            ctile[wave][(r + 8 * half) * 16 + l16] = acc[r];

        wait_async_all();       // my staged portion of the next buffer landed
        __syncthreads();        // ctile visible; next buffer fully staged

        if (lane < 16) {
            const bool same = ((colbase >> 12) == rowdom);
#pragma unroll
            for (int n = 0; n < 16; ++n) {
                float val = ctile[wave][lane * 16 + n];
                if (same) val = 0.0f;            // zero intra-domain block
                float cv = val;
                int   ci = colbase + n;
#pragma unroll
                for (int p = 0; p < KNN; ++p) {  // stable (earliest-index) insert
                    if (cv > bv[p]) {
                        float tv = bv[p]; int ti = bi[p];
                        bv[p] = cv; bi[p] = ci;
                        cv = tv; ci = ti;
                    }
                }
            }
        }
        __syncthreads();        // merge done before ctile is overwritten
    }

    if (lane < 16) {
#pragma unroll
        for (int p = 0; p < KNN; ++p)
            topk[(rowbase + lane) * KNN + p] = bi[p];
    }
}

// ---------------------------------------------------------------------------
// Feature embedding: h0[:, :64] = x[:, :7]@w_num + b_num ; h0[:, 64:] = x[:,7:]@w_cat + b_cat
// ---------------------------------------------------------------------------
__global__ void k_feat_embed(const float* __restrict__ x,
                             const float* __restrict__ w_num, const float* __restrict__ b_num,
                             const float* __restrict__ w_cat, const float* __restrict__ b_cat,
                             float* __restrict__ h0) {
    const int idx = blockIdx.x * blockDim.x + threadIdx.x;
    const int g = idx >> 7;        // global node
    const int j = idx & 127;
    const float* xr = x + g * 10;
    float acc;
    if (j < 64) {
        acc = b_num[j];
#pragma unroll
        for (int p = 0; p < 7; ++p) acc += xr[p] * w_num[p * 64 + j];
    } else {
        const int jj = j - 64;
        acc = b_cat[jj];
#pragma unroll
        for (int p = 0; p < 3; ++p) acc += xr[7 + p] * w_cat[p * 64 + jj];
    }
    h0[g * HDIM + j] = acc;
}

// ---------------------------------------------------------------------------
// Per-domain, per-column batch stats (population variance, ddof=0)
// ---------------------------------------------------------------------------
__global__ void k_colstats(const float* __restrict__ h,
                           float* __restrict__ mean, float* __restrict__ var) {
    const int s = blockIdx.x >> 7;
    const int j = blockIdx.x & 127;
    const float* base = h + (size_t)s * NNODE * HDIM + j;
    float sum = 0.0f, sq = 0.0f;
    for (int i = threadIdx.x; i < NNODE; i += 256) {
        const float x = base[(size_t)i * HDIM];
        sum += x;
        sq  += x * x;
    }
    __shared__ float r1[256], r2[256];
    r1[threadIdx.x] = sum;
    r2[threadIdx.x] = sq;
    __syncthreads();
    for (int off = 128; off > 0; off >>= 1) {
        if (threadIdx.x < off) {
            r1[threadIdx.x] += r1[threadIdx.x + off];
            r2[threadIdx.x] += r2[threadIdx.x + off];
        }
        __syncthreads();
    }
    if (threadIdx.x == 0) {
        const float mu = r1[0] / (float)NNODE;
        mean[blockIdx.x] = mu;
        var[blockIdx.x]  = r2[0] / (float)NNODE - mu * mu;
    }
}

__global__ void k_bn_apply(float* __restrict__ h,
                           const float* __restrict__ mean, const float* __restrict__ var,
                           const float* __restrict__ gamma, const float* __restrict__ beta) {
    const int idx = blockIdx.x * blockDim.x + threadIdx.x;
    const int i = idx >> 7;
    const int s = i >> 12;
    const int j = idx & 127;
    const float mu = mean[s * HDIM + j];
    const float vv = var[s * HDIM + j];
    h[idx] = (h[idx] - mu) * rsqrtf(vv + BN_EPS) * gamma[j] + beta[j];
}

// ---------------------------------------------------------------------------
// Degree / normalization helpers
// ---------------------------------------------------------------------------
__global__ void k_deg_init(float* __restrict__ deg, int n) {
    const int i = blockIdx.x * blockDim.x + threadIdx.x;
    if (i < n) deg[i] = 1.0f;                  // self-loop contribution
}

__global__ void k_deg_accum_enc(const int* __restrict__ ei, float* __restrict__ deg) {
    const int e = blockIdx.x * blockDim.x + threadIdx.x;
    if (e >= S_DOM * NEDGE) return;
    const int s = e / NEDGE;
    const int le = e - s * NEDGE;
    const int dst = ei[s * 2 * NEDGE + NEDGE + le];
    atomicAdd(&deg[s * NNODE + dst], 1.0f);
}

__global__ void k_deg_accum_cross(const int* __restrict__ topk, float* __restrict__ deg) {
    const int e = blockIdx.x * blockDim.x + threadIdx.x;
    if (e >= SN * KNN) return;
    atomicAdd(&deg[topk[e]], 1.0f);
}

__global__ void k_dinv(const float* __restrict__ deg, float* __restrict__ dinv, int n) {
    const int i = blockIdx.x * blockDim.x + threadIdx.x;
    if (i < n) dinv[i] = rsqrtf(fmaxf(deg[i], 1e-12f));
}

// out[i] = bias + dinv[i]^2 * hW[i]  (self-loop term, also initializes buffer)
__global__ void k_gcn_self(const float* __restrict__ hW, const float* __restrict__ dinv,
                           const float* __restrict__ bias, float* __restrict__ out) {
    const int idx = blockIdx.x * blockDim.x + threadIdx.x;
    const int i = idx >> 7;
    const int j = idx & 127;
    const float d = dinv[i];
    out[idx] = bias[j] + d * d * hW[idx];
}

__global__ void k_gcn_scatter_enc(const int* __restrict__ ei, const float* __restrict__ hW,
                                  const float* __restrict__ dinv, float* __restrict__ out) {
    const int idx = blockIdx.x * blockDim.x + threadIdx.x;
    const int j = idx & 127;
    const int e = idx >> 7;
    const int s = e / NEDGE;
    const int le = e - s * NEDGE;
    const int src = ei[s * 2 * NEDGE + le];
    const int dst = ei[s * 2 * NEDGE + NEDGE + le];
    const int gs = s * NNODE + src;
    const int gd = s * NNODE + dst;
    atomicAdd(&out[gd * HDIM + j], dinv[gs] * dinv[gd] * hW[gs * HDIM + j]);
}

__global__ void k_gcn_scatter_cross(const int* __restrict__ topk, const float* __restrict__ hW,
                                    const float* __restrict__ dinv, float* __restrict__ out) {
    const int idx = blockIdx.x * blockDim.x + threadIdx.x;
    const int j = idx & 127;
    const int e = idx >> 7;
    const int i = e / KNN;
    const int c = topk[e];
    atomicAdd(&out[c * HDIM + j], dinv[i] * dinv[c] * hW[i * HDIM + j]);
}

__global__ void k_relu(float* __restrict__ h, int n) {
    const int i = blockIdx.x * blockDim.x + threadIdx.x;
    if (i < n) h[i] = fmaxf(h[i], 0.0f);
}

// ---------------------------------------------------------------------------
// Row L2-normalize: v[i] = emb[i] / max(||emb[i]||, 1e-12)
// ---------------------------------------------------------------------------
__global__ __launch_bounds__(128) void k_rownorm(const float* __restrict__ emb,
                                                 float* __restrict__ v) {
    const int i = blockIdx.x;
    const int j = threadIdx.x;
    const float val = emb[i * HDIM + j];
    __shared__ float red[128];
    red[j] = val * val;
    __syncthreads();
    for (int off = 64; off > 0; off >>= 1) {
        if (j < off) red[j] += red[j + off];
        __syncthreads();
    }
    const float nrm = sqrtf(red[0]);
    v[i * HDIM + j] = val / fmaxf(nrm, 1e-12f);
}

// ---------------------------------------------------------------------------
// Attention fusion of [emb, upd] with a 2-way softmax per node
// ---------------------------------------------------------------------------
__global__ __launch_bounds__(128) void k_attention(const float* __restrict__ emb,
                                                   const float* __restrict__ upd,
                                                   const float* __restrict__ w_att,
                                                   const float* __restrict__ b_att,
                                                   float* __restrict__ out) {
    const int i = blockIdx.x;
    const int j = threadIdx.x;
    const float e0 = emb[i * HDIM + j];
    const float u0 = upd[i * HDIM + j];
    __shared__ float r1[128], r2[128];
    r1[j] = e0 * w_att[j];
    r2[j] = u0 * w_att[j];
    __syncthreads();
    for (int off = 64; off > 0; off >>= 1) {
        if (j < off) { r1[j] += r1[j + off]; r2[j] += r2[j + off]; }
        __syncthreads();
    }
    const float a1 = r1[0] + b_att[0];
    const float a2 = r2[0] + b_att[0];
    const float m  = fmaxf(a1, a2);
    const float x1 = expf(a1 - m);
    const float x2 = expf(a2 - m);
    const float w1 = x1 / (x1 + x2);
    out[i * HDIM + j] = w1 * e0 + (1.0f - w1) * u0;
}

// ---------------------------------------------------------------------------
// Host orchestration
// ---------------------------------------------------------------------------
extern "C" void kernel_launch(void* const* d_in, const int* in_sizes, int n_in,
                              void* d_out, int out_size, void* d_ws, size_t ws_size,
                              hipStream_t stream) {
    const float* src_feature = (const float*)d_in[0];
    const int*   edge_index  = (const int*)d_in[1];
    const float* w_num = (const float*)d_in[2];
    const float* b_num = (const float*)d_in[3];
    const float* w_cat = (const float*)d_in[4];
    const float* b_cat = (const float*)d_in[5];
    const float* w_in  = (const float*)d_in[6];
    const float* b_in  = (const float*)d_in[7];
    const float* gamma = (const float*)d_in[8];
    const float* beta  = (const float*)d_in[9];
    const float* w_g1  = (const float*)d_in[10];
    const float* b_g1  = (const float*)d_in[11];
    const float* w_g2  = (const float*)d_in[12];
    const float* b_g2  = (const float*)d_in[13];
    const float* w_cg  = (const float*)d_in[14];
    const float* b_cg  = (const float*)d_in[15];
    const float* w_att = (const float*)d_in[16];
    const float* b_att = (const float*)d_in[17];
    float* out = (float*)d_out;

    const size_t MAT = (size_t)SN * HDIM;      // 1,572,864 floats
    float* ws   = (float*)d_ws;
    float* h0   = ws;                          // scratch / gcn outputs
    float* h1   = ws + MAT;
    float* hW   = ws + 2 * MAT;
    float* emb  = ws + 3 * MAT;
    float* vbuf = ws + 4 * MAT;
    float* upd  = ws + 5 * MAT;
    float* deg  = ws + 6 * MAT;
    float* dinv = deg + SN;
    float* meanb = dinv + SN;
    float* varb  = meanb + S_DOM * HDIM;
    int*   topk  = (int*)(varb + S_DOM * HDIM);

    const int EW = (int)MAT;                    // SN*128 elementwise work
    const dim3 gemmGrid(SN / 16, HDIM / 16);

    // ---- encoder: feature embed -> linear+relu -> BN ----
    k_feat_embed<<<EW / 256, 256, 0, stream>>>(src_feature, w_num, b_num, w_cat, b_cat, h0);
    k_gemm128<<<gemmGrid, 32, 0, stream>>>(h0, w_in, b_in, 1, h1);
    k_colstats<<<S_DOM * HDIM, 256, 0, stream>>>(h1, meanb, varb);
    k_bn_apply<<<EW / 256, 256, 0, stream>>>(h1, meanb, varb, gamma, beta);

    // degrees (shared by both encoder GCN layers)
    k_deg_init<<<(SN + 255) / 256, 256, 0, stream>>>(deg, SN);
    k_deg_accum_enc<<<(S_DOM * NEDGE) / 256, 256, 0, stream>>>(edge_index, deg);
    k_dinv<<<(SN + 255) / 256, 256, 0, stream>>>(deg, dinv, SN);

    // ---- GCN layer 1 (relu) ----
    k_gemm128<<<gemmGrid, 32, 0, stream>>>(h1, w_g1, nullptr, 0, hW);
    k_gcn_self<<<EW / 256, 256, 0, stream>>>(hW, dinv, b_g1, h0);
    k_gcn_scatter_enc<<<(S_DOM * NEDGE * HDIM) / 256, 256, 0, stream>>>(edge_index, hW, dinv, h0);
    k_relu<<<EW / 256, 256, 0, stream>>>(h0, EW);

    // ---- GCN layer 2 ----
    k_gemm128<<<gemmGrid, 32, 0, stream>>>(h0, w_g2, nullptr, 0, hW);
    k_gcn_self<<<EW / 256, 256, 0, stream>>>(hW, dinv, b_g2, emb);
    k_gcn_scatter_enc<<<(S_DOM * NEDGE * HDIM) / 256, 256, 0, stream>>>(edge_index, hW, dinv, emb);

    // ---- cross-domain: cosine sim (WMMA + async-LDS staging) + masked top-5 ----
    k_rownorm<<<SN, 128, 0, stream>>>(emb, vbuf);
    k_simtopk<<<SN / (16 * TWAVES), 32 * TWAVES, 0, stream>>>(vbuf, topk);

    // ---- cross-source GCN over kNN edges ----
    k_gemm128<<<gemmGrid, 32, 0, stream>>>(emb, w_cg, nullptr, 0, hW);
    k_deg_init<<<(SN + 255) / 256, 256, 0, stream>>>(deg, SN);
    k_deg_accum_cross<<<(SN * KNN + 255) / 256, 256, 0, stream>>>(topk, deg);
    k_dinv<<<(SN + 255) / 256, 256, 0, stream>>>(deg, dinv, SN);
    k_gcn_self<<<EW / 256, 256, 0, stream>>>(hW, dinv, b_cg, upd);
    k_gcn_scatter_cross<<<(SN * KNN * HDIM) / 256, 256, 0, stream>>>(topk, hW, dinv, upd);

    // ---- attention fusion ----
    k_attention<<<SN, 128, 0, stream>>>(emb, upd, w_att, b_att, out);
}